// Net_41008347742647
// MI455X (gfx1250) — compile-verified
//
#include <hip/hip_runtime.h>
#include <math.h>

#define EPSV 1e-06f

typedef float v2f __attribute__((ext_vector_type(2)));
typedef float v8f __attribute__((ext_vector_type(8)));

// ---------------------------------------------------------------------------
// 1) image gradients -> pose0 (N,1,2), a0 (N,1)   N = 16*32*32
// ---------------------------------------------------------------------------
__global__ void k_grad(const float* __restrict__ img, float* __restrict__ pose0,
                       float* __restrict__ a0)
{
    int n = blockIdx.x * blockDim.x + threadIdx.x;
    if (n >= 16 * 32 * 32) return;
    int b = n >> 10, rem = n & 1023, y = rem >> 5, x = rem & 31;
    const float* im = img + b * 1024;
    float gy, gx;
    if (y == 0)       gy = im[1 * 32 + x] - im[0 * 32 + x];
    else if (y == 31) gy = im[31 * 32 + x] - im[30 * 32 + x];
    else              gy = 0.5f * (im[(y + 1) * 32 + x] - im[(y - 1) * 32 + x]);
    if (x == 0)       gx = im[y * 32 + 1] - im[y * 32 + 0];
    else if (x == 31) gx = im[y * 32 + 31] - im[y * 32 + 30];
    else              gx = 0.5f * (im[y * 32 + x + 1] - im[y * 32 + x - 1]);
    float mag = sqrtf(gx * gx + gy * gy);
    pose0[2 * n + 0] = gx / (mag + EPSV);
    pose0[2 * n + 1] = gy / (mag + EPSV);
    a0[n] = mag;
}

// ---------------------------------------------------------------------------
// 2) steerable group conv. One block per output pixel n; phase 1 stages
//    neighbor x-values + bilinear sampling params in LDS; phase 2: one thread
//    per Cout. out = relu(conv + b) * amod   (amod nullable)
//    x: (Bn*H*W, Cin), pose: (Bn*H*W, Cin, 2), Wg: (5,5,Cin,Cout)
// ---------------------------------------------------------------------------
__global__ void k_gconv(const float* __restrict__ x, const float* __restrict__ pose,
                        const float* __restrict__ Wg, const float* __restrict__ bias,
                        const float* __restrict__ amod, float* __restrict__ out,
                        int H, int W, int Cin, int Cout)
{
    extern __shared__ unsigned char smraw[];
    int n  = blockIdx.x;
    int HW = H * W;
    int b  = n / HW, rem = n % HW, y = rem / W, xx = rem % W;
    int items = 25 * Cin;

    float* xn_s = (float*)smraw;
    float* fx_s = xn_s + items;
    float* fy_s = fx_s + items;
    int*   ib_s = (int*)(fy_s + items);

    for (int w = threadIdx.x; w < items; w += blockDim.x) {
        int e = w / Cin, c = w - e * Cin;
        int jy = e / 5, jx = e - jy * 5;
        float dy = (float)(jy - 2), dx = (float)(jx - 2);
        int ny = y + (jy - 2), nx = xx + (jx - 2);
        bool valid = (ny >= 0) & (ny < H) & (nx >= 0) & (nx < W);
        int nyc = ny < 0 ? 0 : (ny > H - 1 ? H - 1 : ny);
        int nxc = nx < 0 ? 0 : (nx > W - 1 ? W - 1 : nx);
        xn_s[w] = valid ? x[(size_t)(b * HW + nyc * W + nxc) * Cin + c] : 0.0f;

        float ux = pose[((size_t)n * Cin + c) * 2 + 0];
        float uy = pose[((size_t)n * Cin + c) * 2 + 1];
        float rx =  ux * dx + uy * dy;
        float ry = -uy * dx + ux * dy;
        float px = fminf(fmaxf(rx + 2.0f, 0.0f), 4.0f);
        float py = fminf(fmaxf(ry + 2.0f, 0.0f), 4.0f);
        float xf = fminf(floorf(px), 3.0f);
        float yf = fminf(floorf(py), 3.0f);
        fx_s[w] = px - xf;
        fy_s[w] = py - yf;
        ib_s[w] = (((int)yf * 5 + (int)xf) * Cin + c) * Cout;
    }
    __syncthreads();

    int d = threadIdx.x;
    if (d < Cout) {
        float acc    = bias[d];
        int   step_x = Cin * Cout;
        int   step_y = 5 * Cin * Cout;
        for (int w = 0; w < items; ++w) {
            float xv = xn_s[w];
            int base = ib_s[w] + d;
            float fx = fx_s[w], fy = fy_s[w];
            float w00 = Wg[base];
            float w01 = Wg[base + step_x];
            float w10 = Wg[base + step_y];
            float w11 = Wg[base + step_y + step_x];
            float wint = w00 * (1.0f - fx) * (1.0f - fy) + w01 * fx * (1.0f - fy)
                       + w10 * (1.0f - fx) * fy          + w11 * fx * fy;
            acc = fmaf(xv, wint, acc);
        }
        acc = fmaxf(acc, 0.0f);
        if (amod) acc *= amod[(size_t)n * Cout + d];
        out[(size_t)n * Cout + d] = acc;
    }
}

// ---------------------------------------------------------------------------
// 3) capsule pool. One block per pooled position. M = 4*Cin <= 256, M*f <= 256.
// ---------------------------------------------------------------------------
__global__ void k_pool(const float* __restrict__ x, const float* __restrict__ a,
                       const float* __restrict__ pose, const float* __restrict__ t,
                       const float* __restrict__ wx,
                       float* __restrict__ xout, float* __restrict__ aout,
                       float* __restrict__ pout, float* __restrict__ frout,
                       int H, int W, int Cin, int f, int Cout)
{
    __shared__ float ab_s[256];
    __shared__ float pb_s[512];
    __shared__ float xb_s[256];

    int n  = blockIdx.x;
    int Ho = H / 2, Wo = W / 2;
    int b  = n / (Ho * Wo), rem = n % (Ho * Wo), ho = rem / Wo, wo = rem % Wo;
    int M  = 4 * Cin;

    for (int m = threadIdx.x; m < M; m += blockDim.x) {
        int q = m / Cin, cin = m - q * Cin;
        int dy = q >> 1, dx = q & 1;
        int pix = (b * H + (2 * ho + dy)) * W + (2 * wo + dx);
        ab_s[m] = fmaxf(a[(size_t)pix * Cin + cin], 0.0f);   // relu(a)
        pb_s[2 * m + 0] = pose[((size_t)pix * Cin + cin) * 2 + 0];
        pb_s[2 * m + 1] = pose[((size_t)pix * Cin + cin) * 2 + 1];
        for (int j = 0; j < f; ++j)
            xb_s[m * f + j] = x[((size_t)pix * Cin + cin) * f + j];
    }
    __syncthreads();

    int o = threadIdx.x;
    if (o < Cout) {
        float absum = 0.0f, Sx = 0.0f, Sy = 0.0f;
        for (int m = 0; m < M; ++m) {
            float tx = t[((size_t)m * Cout + o) * 2 + 0];
            float ty = t[((size_t)m * Cout + o) * 2 + 1];
            float inv = 1.0f / (sqrtf(tx * tx + ty * ty) + EPSV);
            tx *= inv; ty *= inv;
            float p0 = pb_s[2 * m], p1 = pb_s[2 * m + 1];
            float vx = p0 * tx - p1 * ty;
            float vy = p0 * ty + p1 * tx;
            float ab = ab_s[m];
            absum += ab;
            Sx = fmaf(ab, vx, Sx);
            Sy = fmaf(ab, vy, Sy);
        }
        float Sn = sqrtf(Sx * Sx + Sy * Sy) + EPSV;
        float px_ = Sx / Sn, py_ = Sy / Sn;

        float accA = 0.0f, accX = 0.0f;
        for (int m = 0; m < M; ++m) {
            float tx = t[((size_t)m * Cout + o) * 2 + 0];
            float ty = t[((size_t)m * Cout + o) * 2 + 1];
            float inv = 1.0f / (sqrtf(tx * tx + ty * ty) + EPSV);
            tx *= inv; ty *= inv;
            float p0 = pb_s[2 * m], p1 = pb_s[2 * m + 1];
            float vx = p0 * tx - p1 * ty;
            float vy = p0 * ty + p1 * tx;
            float sig = vx * px_ + vy * py_;
            float feat = 0.0f;
            for (int j = 0; j < f; ++j)
                feat = fmaf(xb_s[m * f + j], wx[((size_t)m * f + j) * Cout + o], feat);
            float ab = ab_s[m];
            accA = fmaf(ab, sig, accA);
            accX = fmaf(ab * sig, feat, accX);
        }
        xout[(size_t)n * Cout + o] = accX;
        aout[(size_t)n * Cout + o] = accA / (absum + EPSV);
        pout[((size_t)n * Cout + o) * 2 + 0] = px_;
        pout[((size_t)n * Cout + o) * 2 + 1] = py_;
    }

    if (frout) {  // only needed for the final pool
        int CF = Cin * f;
        for (int j2 = threadIdx.x; j2 < CF; j2 += blockDim.x) {
            int cin = j2 / f, jf = j2 - cin * f;
            float s = 0.0f;
            for (int q = 0; q < 4; ++q)
                s = fmaf(ab_s[q * Cin + cin], xb_s[(q * Cin + cin) * f + jf], s);
            frout[(size_t)n * CF + j2] = s;
        }
    }
}

// ---------------------------------------------------------------------------
// 4a) rec_in = concat(fr(16,64), a5(16,10), pose5(16,20)) padded to (16,96)
// ---------------------------------------------------------------------------
__global__ void k_recin(const float* __restrict__ fr, const float* __restrict__ a5,
                        const float* __restrict__ p5, float* __restrict__ rec_in)
{
    int i = blockIdx.x * blockDim.x + threadIdx.x;
    if (i >= 16 * 96) return;
    int bb = i / 96, j = i - bb * 96;
    float v;
    if (j < 64)      v = fr[bb * 64 + j];
    else if (j < 74) v = a5[bb * 10 + (j - 64)];
    else if (j < 94) v = p5[bb * 20 + (j - 74)];
    else             v = 0.0f;
    rec_in[i] = v;
}

// ---------------------------------------------------------------------------
// 4b) zero-pad rW1 (94,512) -> (96,512) so the GEMM needs no K guard.
//     (rec_in cols 94,95 are zero, so padded rows are inert.)
// ---------------------------------------------------------------------------
__global__ void k_padW1(const float* __restrict__ w, float* __restrict__ wp)
{
    int i = blockIdx.x * blockDim.x + threadIdx.x;
    if (i >= 96 * 512) return;
    int k = i >> 9;                       // row (N = 512 = 2^9)
    wp[i] = (k < 94) ? w[i] : 0.0f;
}

// ---------------------------------------------------------------------------
// 5) WMMA GEMM:  out(16,N) = act( A(16,KP) @ W(KP,N) + bias )
//    One wave per 16-column tile. v_wmma_f32_16x16x4_f32, K stepped by 4,
//    branch-free, unrolled. ISA lane layout:
//      A vgpr0/1 = K={0,1} (lanes 0-15) / K={2,3} (lanes 16-31), M = lane%16
//      B vgpr0/1 = K rows {0,1}/{2,3} split across lane halves, N = lane%16
//      D: vgpr v -> row v + 8*(lane>=16), col = lane%16
//    ACT: 0 = relu, 1 = sigmoid.
// ---------------------------------------------------------------------------
template <int KP, int ACT>
__global__ void k_gemm_wmma(const float* __restrict__ A, const float* __restrict__ Wg,
                            const float* __restrict__ bias, float* __restrict__ out,
                            int N)
{
    int tile = blockIdx.x;
    int lane = threadIdx.x;          // 0..31, wave32
    int lo16 = lane & 15;
    int hi   = lane >> 4;            // 0 or 1
    int col  = tile * 16 + lo16;

    const float* ap = A  + (size_t)lo16 * KP + 2 * hi;   // this lane's A pair
    const float* bp = Wg + (size_t)(2 * hi) * N + col;   // this lane's B rows

    v8f c = {};
#pragma unroll 8
    for (int k = 0; k < KP; k += 4) {
        v2f av, bv;
        av.x = ap[0];
        av.y = ap[1];
        bv.x = bp[0];
        bv.y = bp[(size_t)N];
        c = __builtin_amdgcn_wmma_f32_16x16x4_f32(false, av, false, bv,
                                                  (short)0, c, false, false);
        ap += 4;
        bp += (size_t)4 * N;
    }

    float bcol = bias[col];
#pragma unroll
    for (int v = 0; v < 8; ++v) {
        int row = v + 8 * hi;
        float val = c[v] + bcol;
        val = (ACT == 0) ? fmaxf(val, 0.0f) : 1.0f / (1.0f + expf(-val));
        out[(size_t)row * N + col] = val;
    }
}

// ---------------------------------------------------------------------------
// 6) heads: log_softmax(x5) -> out[0:160], a5 -> out[160:320], p5 -> out[320:640]
// ---------------------------------------------------------------------------
__global__ void k_heads(const float* __restrict__ x5, const float* __restrict__ a5,
                        const float* __restrict__ p5, float* __restrict__ out)
{
    int tb = threadIdx.x;
    if (tb < 16) {
        const float* xr = x5 + tb * 10;
        float mx = xr[0];
        for (int i = 1; i < 10; ++i) mx = fmaxf(mx, xr[i]);
        float s = 0.0f;
        for (int i = 0; i < 10; ++i) s += expf(xr[i] - mx);
        float lse = logf(s) + mx;
        for (int i = 0; i < 10; ++i) out[tb * 10 + i] = xr[i] - lse;
        for (int i = 0; i < 10; ++i) out[160 + tb * 10 + i] = a5[tb * 10 + i];
        for (int i = 0; i < 20; ++i) out[320 + tb * 20 + i] = p5[tb * 20 + i];
    }
}

// ---------------------------------------------------------------------------
extern "C" void kernel_launch(void* const* d_in, const int* in_sizes, int n_in,
                              void* d_out, int out_size, void* d_ws, size_t ws_size,
                              hipStream_t stream)
{
    const float* img = (const float*)d_in[0];
    const float* cw1 = (const float*)d_in[1];  const float* cb1 = (const float*)d_in[2];
    const float* cw2 = (const float*)d_in[3];  const float* cb2 = (const float*)d_in[4];
    const float* cw3 = (const float*)d_in[5];  const float* cb3 = (const float*)d_in[6];
    const float* cw4 = (const float*)d_in[7];  const float* cb4 = (const float*)d_in[8];
    const float* cw5 = (const float*)d_in[9];  const float* cb5 = (const float*)d_in[10];
    const float* pt1 = (const float*)d_in[11]; const float* px1 = (const float*)d_in[12];
    const float* pt2 = (const float*)d_in[13]; const float* px2 = (const float*)d_in[14];
    const float* pt3 = (const float*)d_in[15]; const float* px3 = (const float*)d_in[16];
    const float* pt4 = (const float*)d_in[17]; const float* px4 = (const float*)d_in[18];
    const float* pt5 = (const float*)d_in[19]; const float* px5 = (const float*)d_in[20];
    const float* rW1 = (const float*)d_in[21]; const float* rb1 = (const float*)d_in[22];
    const float* rW2 = (const float*)d_in[23]; const float* rb2 = (const float*)d_in[24];
    const float* rW3 = (const float*)d_in[25]; const float* rb3 = (const float*)d_in[26];

    float* out = (float*)d_out;
    float* ws  = (float*)d_ws;
    size_t off = 0;
    auto A = [&](size_t nflt) { float* p = ws + off; off += nflt; return p; };

    float* pose0 = A(32768);  float* a0  = A(16384);
    float* x1    = A(262144);
    float* p1x   = A(65536);  float* p1a = A(65536);  float* p1p = A(131072);
    float* x2    = A(65536);
    float* p2x   = A(32768);  float* p2a = A(32768);  float* p2p = A(65536);
    float* x3    = A(32768);
    float* p3x   = A(8192);   float* p3a = A(8192);   float* p3p = A(16384);
    float* x4    = A(8192);
    float* p4x   = A(4096);   float* p4a = A(4096);   float* p4p = A(8192);
    float* x5    = A(4096);
    float* p5x   = A(160);    float* p5a = A(160);    float* p5p = A(320);
    float* fr    = A(1024);
    float* recin = A(1536);   float* h1  = A(8192);   float* h2  = A(16384);
    float* w1p   = A(49152);  // padded rW1: 96 x 512

    // stage 0: gradients
    k_grad<<<(16384 + 255) / 256, 256, 0, stream>>>(img, pose0, a0);

    // conv1 (Cin=1,Cout=16, 32x32) -> pool1 (Cin=1,f=16,Cout=16)
    k_gconv<<<16384, 64, 25 * 1 * 16, stream>>>(img, pose0, cw1, cb1, nullptr, x1, 32, 32, 1, 16);
    k_pool <<<4096, 64, 0, stream>>>(x1, a0, pose0, pt1, px1, p1x, p1a, p1p, nullptr, 32, 32, 1, 16, 16);

    // conv2 (16->16, 16x16) -> pool2 (Cin=16,f=1,Cout=32)
    k_gconv<<<4096, 64, 25 * 16 * 16, stream>>>(p1x, p1p, cw2, cb2, p1a, x2, 16, 16, 16, 16);
    k_pool <<<1024, 64, 0, stream>>>(x2, p1a, p1p, pt2, px2, p2x, p2a, p2p, nullptr, 16, 16, 16, 1, 32);

    // conv3 (32->32, 8x8) -> pool3 (Cin=32,f=1,Cout=32)
    k_gconv<<<1024, 64, 25 * 32 * 16, stream>>>(p2x, p2p, cw3, cb3, p2a, x3, 8, 8, 32, 32);
    k_pool <<<256, 64, 0, stream>>>(x3, p2a, p2p, pt3, px3, p3x, p3a, p3p, nullptr, 8, 8, 32, 1, 32);

    // conv4 (32->32, 4x4) -> pool4 (Cin=32,f=1,Cout=64)
    k_gconv<<<256, 64, 25 * 32 * 16, stream>>>(p3x, p3p, cw4, cb4, p3a, x4, 4, 4, 32, 32);
    k_pool <<<64, 64, 0, stream>>>(x4, p3a, p3p, pt4, px4, p4x, p4a, p4p, nullptr, 4, 4, 32, 1, 64);

    // conv5 (64->64, 2x2) -> pool5 (Cin=64,f=1,Cout=10) with fr
    k_gconv<<<64, 64, 25 * 64 * 16, stream>>>(p4x, p4p, cw5, cb5, p4a, x5, 2, 2, 64, 64);
    k_pool <<<16, 64, 0, stream>>>(x5, p4a, p4p, pt5, px5, p5x, p5a, p5p, fr, 2, 2, 64, 1, 10);

    // reconstruction MLP on tensor cores (M=16 == batch)
    k_recin<<<(16 * 96 + 255) / 256, 256, 0, stream>>>(fr, p5a, p5p, recin);
    k_padW1<<<(96 * 512 + 255) / 256, 256, 0, stream>>>(rW1, w1p);
    k_gemm_wmma<96,   0><<<512 / 16,  32, 0, stream>>>(recin, w1p, rb1, h1,        512);
    k_gemm_wmma<512,  0><<<1024 / 16, 32, 0, stream>>>(h1,    rW2, rb2, h2,        1024);
    k_gemm_wmma<1024, 1><<<784 / 16,  32, 0, stream>>>(h2,    rW3, rb3, out + 640, 784);

    // heads: log_softmax + a + pose
    k_heads<<<1, 32, 0, stream>>>(p5x, p5a, p5p, out);

    (void)in_sizes; (void)n_in; (void)out_size; (void)ws_size;
}